// SimilarityRedistributor_7911329760049
// MI455X (gfx1250) — compile-verified
//
#include <hip/hip_runtime.h>

// COO SpMV scatter-add for MI455X (gfx1250, wave32).
// out = segment_sum(S_vals * logits[S_cols], S_rows, V) - ALPHA * logits
//
// Memory-bound: 98 MB streamed COO data (NT hints), 512 KB L2-resident gather
// table, native f32 L2 atomics for the scatter. No dense tile structure ->
// WMMA not applicable; optimize the data-movement path instead.

typedef int   v4i __attribute__((ext_vector_type(4)));
typedef float v4f __attribute__((ext_vector_type(4)));

#define SR_ALPHA 0.1f

__device__ __forceinline__ void sr_atomic_add(float* p, float v) {
    // Relaxed, device-scope float add. The clang::atomic attribute attaches
    // !amdgpu.no.fine.grained.memory / !amdgpu.ignore.denormal.mode metadata
    // so the backend selects hardware global_atomic_add_f32 (no CAS loop).
#if defined(__has_cpp_attribute) && __has_cpp_attribute(clang::atomic)
    [[clang::atomic(no_remote_memory, no_fine_grained_memory, ignore_denormal_mode)]] {
        __hip_atomic_fetch_add(p, v, __ATOMIC_RELAXED, __HIP_MEMORY_SCOPE_AGENT);
    }
#else
    __hip_atomic_fetch_add(p, v, __ATOMIC_RELAXED, __HIP_MEMORY_SCOPE_AGENT);
#endif
}

// Defined FIRST in the file so the disasm snippet shows this kernel:
// verify global_atomic_add_f32 + nontemporal (TH=NT) b128 loads.
__global__ __launch_bounds__(256) void sr_scatter_kernel(const float* __restrict__ logits,
                                                         const int* __restrict__ rows,
                                                         const int* __restrict__ cols,
                                                         const float* __restrict__ vals,
                                                         float* __restrict__ out,
                                                         int nnz) {
    int t    = blockIdx.x * 256 + threadIdx.x;
    int base = t * 4;

    if (base + 3 < nnz) {
        // 16B coalesced nontemporal loads: these arrays are read exactly once;
        // keep them out of the way of the L2-resident logits table.
        v4i r = __builtin_nontemporal_load((const v4i*)(rows + base));
        v4i c = __builtin_nontemporal_load((const v4i*)(cols + base));
        v4f v = __builtin_nontemporal_load((const v4f*)(vals + base));

        // Gather from 512KB logits table (L2-resident), scale.
        float g0 = logits[c[0]] * v[0];
        float g1 = logits[c[1]] * v[1];
        float g2 = logits[c[2]] * v[2];
        float g3 = logits[c[3]] * v[3];

        sr_atomic_add(&out[r[0]], g0);
        sr_atomic_add(&out[r[1]], g1);
        sr_atomic_add(&out[r[2]], g2);
        sr_atomic_add(&out[r[3]], g3);
    } else if (base < nnz) {
        // Tail (not hit for NNZ = 8,192,000, but keep it generic).
        for (int i = base; i < nnz; ++i) {
            float g = logits[cols[i]] * vals[i];
            sr_atomic_add(&out[rows[i]], g);
        }
    }
}

__global__ __launch_bounds__(256) void sr_init_kernel(const float* __restrict__ logits,
                                                      float* __restrict__ out,
                                                      int V) {
    int i = blockIdx.x * 256 + threadIdx.x;
    if (i < V) {
        out[i] = -SR_ALPHA * logits[i];
    }
}

extern "C" void kernel_launch(void* const* d_in, const int* in_sizes, int n_in,
                              void* d_out, int out_size, void* d_ws, size_t ws_size,
                              hipStream_t stream) {
    const float* logits = (const float*)d_in[0];
    const int*   rows   = (const int*)d_in[1];
    const int*   cols   = (const int*)d_in[2];
    const float* vals   = (const float*)d_in[3];
    float*       out    = (float*)d_out;

    const int V   = in_sizes[0];
    const int nnz = in_sizes[1];

    // Pass 1: out = -alpha * logits (also initializes poisoned output).
    int init_blocks = (V + 255) / 256;
    sr_init_kernel<<<init_blocks, 256, 0, stream>>>(logits, out, V);

    // Pass 2: scatter-add, 4 nnz per thread (B128 loads), stream-ordered
    // after the init pass.
    int threads = (nnz + 3) / 4;
    int blocks  = (threads + 255) / 256;
    sr_scatter_kernel<<<blocks, 256, 0, stream>>>(logits, rows, cols, vals, out, nnz);
}